// KNearestNeighborAttention_20813411516859
// MI455X (gfx1250) — compile-verified
//
#include <hip/hip_runtime.h>
#include <math.h>

// KNN attention: B=4, N=4096, D=256, K=16 (f32 in / f32 out).
// Phase 1: distance tiles via V_WMMA_F32_16X16X4_F32 (xyz padded to K=4).
//          Store (|c|^2 - 2*dot) to LDS (row stride 18, conflict-free b64 reads),
//          register-resident sorted top-17 per query, indices -> d_ws.
// Phase 2: gather + scores + softmax + weighted sum, one wave per query.

#define D_MODEL 256
#define KNB 16
#define NQ_PER_BLOCK 32   // queries per wave (two 16-row WMMA tiles)
#define RSTRIDE 18        // LDS row stride in floats: 72B, 8B-aligned, conflict-free

typedef __attribute__((ext_vector_type(2))) float v2f;
typedef __attribute__((ext_vector_type(8))) float v8f;

__global__ __launch_bounds__(32)
void knn_search_kernel(const float* __restrict__ xyz, int* __restrict__ knn,
                       int B, int N)
{
    __shared__ __align__(16) float s_dot[32 * RSTRIDE]; // 32 rows x 16 cols of (|c|^2 - 2*dot)

    const int lane   = threadIdx.x;          // 0..31
    const int qtiles = N / NQ_PER_BLOCK;
    const int b      = blockIdx.x / qtiles;
    const int qbase  = (blockIdx.x % qtiles) * NQ_PER_BLOCK;

    const int  l15 = lane & 15;
    const bool hi  = lane >= 16;

    // Load the two query points this lane contributes to (A0: rows 0-15, A1: rows 16-31).
    const float* q0p = xyz + (size_t)(b * N + qbase + l15) * 3;
    const float* q1p = xyz + (size_t)(b * N + qbase + 16 + l15) * 3;
    const float q0x = q0p[0], q0y = q0p[1], q0z = q0p[2];
    const float q1x = q1p[0], q1y = q1p[1], q1z = q1p[2];

    // A-fragment layout (16x4 f32): lanes 0-15 hold (K=0,K=1)=(x,y); lanes 16-31 hold (K=2,K=3)=(z,0).
    v2f a0, a1;
    a0.x = hi ? q0z : q0x;  a0.y = hi ? 0.0f : q0y;
    a1.x = hi ? q1z : q1x;  a1.y = hi ? 0.0f : q1y;

    // |q|^2 for the query row this lane owns during selection (row == lane).
    const float ox = hi ? q1x : q0x, oy = hi ? q1y : q0y, oz = hi ? q1z : q0z;
    const float sqq = ox * ox + oy * oy + oz * oz;

    // Sorted ascending top-17 (dist, idx) in registers.
    float nd[17]; int ni[17];
    #pragma unroll
    for (int j = 0; j < 17; ++j) { nd[j] = 3.4e38f; ni[j] = 0; }

    const int ctiles = N / 16;
    for (int t = 0; t < ctiles; ++t) {
        const float* cp = xyz + (size_t)(b * N + t * 16 + l15) * 3;
        // Prefetch next candidate tile (global_prefetch_b8).
        if (t + 1 < ctiles) __builtin_prefetch((const void*)(cp + 48), 0, 1);
        const float cx = cp[0], cy = cp[1], cz = cp[2];
        const float sqc = cx * cx + cy * cy + cz * cz;   // |c|^2 of column l15

        // B-fragment (4x16 f32): lanes 0-15 -> (x,y); lanes 16-31 -> (z,0); N = lane&15.
        v2f bfrag; bfrag.x = hi ? cz : cx; bfrag.y = hi ? 0.0f : cy;

        v8f zero = {};
        // dot tiles: rows = queries, cols = candidates
        v8f d0 = __builtin_amdgcn_wmma_f32_16x16x4_f32(false, a0, false, bfrag,
                                                       (short)0, zero, false, false);
        v8f d1 = __builtin_amdgcn_wmma_f32_16x16x4_f32(false, a1, false, bfrag,
                                                       (short)0, zero, false, false);

        // C/D layout: VGPR r -> row r (lanes 0-15) / row r+8 (lanes 16-31), col = lane&15.
        // Fold |c|^2 here so selection needs one LDS read + one add per candidate.
        const int rofs = hi ? 8 : 0;
        #pragma unroll
        for (int r = 0; r < 8; ++r)
            s_dot[(r + rofs) * RSTRIDE + l15]      = fmaf(-2.0f, d0[r], sqc);
        #pragma unroll
        for (int r = 0; r < 8; ++r)
            s_dot[(16 + r + rofs) * RSTRIDE + l15] = fmaf(-2.0f, d1[r], sqc);
        __syncthreads();

        // Selection: lane q == query row q. 8x ds_load_b64 (stride-18 rows: conflict-free).
        float vals[16];
        {
            const float2* rowp = (const float2*)&s_dot[lane * RSTRIDE];
            #pragma unroll
            for (int c2 = 0; c2 < 8; ++c2) {
                const float2 v = rowp[c2];
                vals[2 * c2]     = v.x;
                vals[2 * c2 + 1] = v.y;
            }
        }
        #pragma unroll
        for (int c = 0; c < 16; ++c) {
            const float d2 = sqq + vals[c];
            if (d2 < nd[16]) {
                float dcur = d2; int icur = t * 16 + c;
                #pragma unroll
                for (int j = 0; j < 17; ++j) {
                    const bool  sw = dcur < nd[j];   // strict <: stable, lower index wins ties
                    const float td = nd[j]; const int ti = ni[j];
                    nd[j] = sw ? dcur : td;  ni[j] = sw ? icur : ti;
                    dcur  = sw ? td   : dcur; icur = sw ? ti  : icur;
                }
            }
        }
        __syncthreads();
    }

    // Drop slot 0 (self), emit neighbors 1..16.
    int* op = knn + (size_t)(b * N + qbase + lane) * KNB;
    #pragma unroll
    for (int j = 0; j < KNB; ++j) op[j] = ni[j + 1];
}

__global__ __launch_bounds__(128)
void knn_attn_kernel(const float* __restrict__ feats, const int* __restrict__ knn,
                     float* __restrict__ out, int B, int N)
{
    __shared__ float s_q[4][D_MODEL];
    __shared__ float s_w[4][KNB];
    __shared__ int   s_i[4][KNB];

    const int lane  = threadIdx.x & 31;
    const int wv    = threadIdx.x >> 5;
    const int qglob = blockIdx.x * 4 + wv;         // 0 .. B*N-1
    const int b     = qglob / N;
    const int n     = qglob % N;

    const float* fbase = feats + (size_t)b * N * D_MODEL;

    // Stage query features (256 f32) into LDS; fetch neighbor indices.
    {
        const float4* q4 = (const float4*)(fbase + (size_t)n * D_MODEL);
        float4 v0 = q4[lane * 2], v1 = q4[lane * 2 + 1];
        float4* s4 = (float4*)s_q[wv];
        s4[lane * 2] = v0; s4[lane * 2 + 1] = v1;
    }
    if (lane < KNB) s_i[wv][lane] = knn[(size_t)qglob * KNB + lane];
    __syncthreads();

    // Scores: lane pair (l, l+16) splits the 256-dim dot for neighbor l&15.
    const int   kn   = lane & 15;
    const int   half = lane >> 4;
    const int   gidx = s_i[wv][kn];
    const float4* g4  = (const float4*)(fbase + (size_t)gidx * D_MODEL);
    const float4* sq4 = (const float4*)s_q[wv];
    float acc = 0.0f;
    #pragma unroll 8
    for (int c = half * 32; c < half * 32 + 32; ++c) {
        const float4 g = g4[c]; const float4 q = sq4[c];
        acc += g.x * q.x + g.y * q.y + g.z * q.z + g.w * q.w;
    }
    acc += __shfl_xor(acc, 16);
    const float score = acc * 0.0625f;             // 1/sqrt(256)

    // Softmax over 16 neighbors (halves are mirrored; xor {8,4,2,1} stays in-half).
    float m = score;
    #pragma unroll
    for (int o = 8; o >= 1; o >>= 1) m = fmaxf(m, __shfl_xor(m, o));
    float w = __expf(score - m);
    float s = w;
    #pragma unroll
    for (int o = 8; o >= 1; o >>= 1) s += __shfl_xor(s, o);
    w /= s;
    if (lane < KNB) s_w[wv][lane] = w;
    __syncthreads();

    // Output: lane owns channels [lane*8, lane*8+8).
    float4 o0 = {0, 0, 0, 0}, o1 = {0, 0, 0, 0};
    #pragma unroll
    for (int k = 0; k < KNB; ++k) {
        const float wk = s_w[wv][k];
        const float4* gk = (const float4*)(fbase + (size_t)s_i[wv][k] * D_MODEL);
        const float4 g0 = gk[lane * 2], g1 = gk[lane * 2 + 1];
        o0.x += wk * g0.x; o0.y += wk * g0.y; o0.z += wk * g0.z; o0.w += wk * g0.w;
        o1.x += wk * g1.x; o1.y += wk * g1.y; o1.z += wk * g1.z; o1.w += wk * g1.w;
    }
    float4* op = (float4*)(out + (size_t)qglob * D_MODEL);
    op[lane * 2] = o0; op[lane * 2 + 1] = o1;
}

extern "C" void kernel_launch(void* const* d_in, const int* in_sizes, int n_in,
                              void* d_out, int out_size, void* d_ws, size_t ws_size,
                              hipStream_t stream) {
    const float* xyz   = (const float*)d_in[0];   // (B, N, 3) f32
    const float* feats = (const float*)d_in[1];   // (B, N, 256) f32
    float*       outp  = (float*)d_out;           // (B, N, 256) f32

    const int BN = in_sizes[0] / 3;
    const int N  = 4096;                          // reference setup
    const int B  = BN / N;

    int* knn = (int*)d_ws;                        // B*N*16 ints = 1 MB scratch

    knn_search_kernel<<<B * (N / NQ_PER_BLOCK), 32, 0, stream>>>(xyz, knn, B, N);
    knn_attn_kernel<<<(B * N) / 4, 128, 0, stream>>>(feats, knn, outp, B, N);
}